// OrthogonalLayer2D_66133906424505
// MI455X (gfx1250) — compile-verified
//
#include <hip/hip_runtime.h>
#include <hip/hip_bf16.h>

typedef __attribute__((ext_vector_type(2))) float v2f;
typedef __attribute__((ext_vector_type(8))) float v8f;

#define NB 8                      // basis vectors per group (N)
#define DLEN 1024                 // vector length (H*W)
#define SROW 1044                 // LDS row stride in floats; 1044 % 64 == 20 (conflict-free)
#define NSTRIDE (32 * 128 * 1024) // element stride between n-slices of x

// Two (m,c) groups per block: LDS rows 0..7 = group 2b, rows 8..15 = group 2b+1.
__global__ __launch_bounds__(256)
void ortho_gs_wmma2_kernel(const float* __restrict__ x, float* __restrict__ out)
{
    __shared__ float Vsh[16 * SROW];   // 16 x 1024 packed tile (padded rows)
    __shared__ float PartG[8][128];    // per-wave partial Gram (two 8x8 blocks)
    __shared__ float Gsh[128];         // reduced Gram: [0..63]=grp0, [64..127]=grp1
    __shared__ float Wsh[256];         // 16x16 block-diagonal W = L^{-1}

    const int  t    = threadIdx.x;
    const int  lane = t & 31;
    const int  wv   = t >> 5;
    const long g0   = (long)blockIdx.x * 2;   // first group of the pair

    // ---- load: per n-slice, 2048 contiguous floats (grp0|grp1), float4 ----
    const float* src = x + g0 * (long)DLEN;
#pragma unroll
    for (int r = 0; r < NB; ++r) {
#pragma unroll
        for (int j = 0; j < 2; ++j) {
            const int   f    = j * 256 + t;          // float4 index 0..511
            const int   col4 = f * 4;                // 0..2047
            const int   row  = r + ((col4 >= DLEN) ? 8 : 0);
            const int   lcol = col4 & (DLEN - 1);
            const float4 v = *(const float4*)(src + (long)r * NSTRIDE + col4);
            *(float4*)(&Vsh[row * SROW + lcol]) = v;
        }
    }
    __syncthreads();

    // ---- phase 1: 16x16 G = X X^T via v_wmma_f32_16x16x4_f32 (no masking) ----
    const int m    = lane & 15;            // A: M = lane&15 (all 16 rows valid)
    const int koff = (lane >> 4) * 2;      // A: K pair by lane>>4 (== B's ks)

    v8f acc = {};
    const int kbase = wv * 128;            // each wave owns a 128-wide K slice
    for (int kt = 0; kt < 32; ++kt) {
        const int kc = kbase + kt * 4;
        v2f a;                              // identical lane pattern for A and B
        a.x = Vsh[m * SROW + kc + koff];
        a.y = Vsh[m * SROW + kc + koff + 1];
        acc = __builtin_amdgcn_wmma_f32_16x16x4_f32(false, a, false, a,
                                                    (short)0, acc, false, false);
    }
    // keep the two 8x8 blocks in columns N<8:
    //   grp0: lanes 0..7  (M=r,   N=lane);  grp1: lanes 16..23 (M=r+8, N=lane-16)
    if ((lane & 15) < 8) {
        const int grp = lane >> 4;
#pragma unroll
        for (int r = 0; r < NB; ++r)
            PartG[wv][grp * 64 + r * 8 + (lane & 15)] = acc[r];
    }
    __syncthreads();
    if (t < 128) {
        float s = 0.0f;
#pragma unroll
        for (int w = 0; w < 8; ++w) s += PartG[w][t];
        Gsh[t] = s;
    }
    Wsh[t] = 0.0f;                         // zero full 16x16 W (256 threads)
    __syncthreads();

    // ---- two 8x8 Choleskys + triangular inverses (wave0/wave1, lane 0) ----
    if (lane == 0 && wv < 2) {
        const int    grp = wv;
        const float* G   = &Gsh[grp * 64];
        float L[8][8];
        for (int i = 0; i < 8; ++i)
            for (int j = 0; j <= i; ++j) {
                float s = G[i * 8 + j];
                for (int k = 0; k < j; ++k) s -= L[i][k] * L[j][k];
                if (i == j) L[i][i] = (s > 0.0f) ? sqrtf(s) : 0.0f;
                else        L[i][j] = (L[j][j] > 0.0f) ? s / L[j][j] : 0.0f;
            }
        float W[8][8];
        for (int i = 0; i < 8; ++i) {
            const float dinv = (L[i][i] > 0.0f) ? 1.0f / L[i][i] : 0.0f;
            for (int j = 0; j < 8; ++j) W[i][j] = 0.0f;
            W[i][i] = dinv;                // dinv==0 zeroes the row (NaN->0)
            for (int j = 0; j < i; ++j) {
                float s = 0.0f;
                for (int k = j; k < i; ++k) s += L[i][k] * W[k][j];
                W[i][j] = -s * dinv;
            }
        }
        const int o = grp * 8;             // place into block-diagonal position
        for (int i = 0; i < 8; ++i)
            for (int j = 0; j < 8; ++j)
                Wsh[(o + i) * 16 + (o + j)] = W[i][j];
    }
    __syncthreads();

    // ---- phase 2: B = W * X (16x16 x 16x1024), K=16 as 4 WMMAs, in-place ----
    v2f aop[4];                            // A = W rows, K chunks {0,4,8,12}
#pragma unroll
    for (int q = 0; q < 4; ++q) {
        aop[q].x = Wsh[m * 16 + q * 4 + koff];
        aop[q].y = Wsh[m * 16 + q * 4 + koff + 1];
    }

    const int n      = lane & 15;          // B(4x16): N = lane&15
    const int ks     = (lane >> 4) * 2;    // B: K pair by lane>>4
    const int rowoff = (lane >> 4) * 8;    // D: lanes 16..31 hold M=r+8
    for (int tile = 0; tile < 8; ++tile) {
        const int col0 = (wv * 8 + tile) * 16;
        v8f d = {};
#pragma unroll
        for (int q = 0; q < 4; ++q) {
            v2f b;
            b.x = Vsh[(q * 4 + ks    ) * SROW + col0 + n];
            b.y = Vsh[(q * 4 + ks + 1) * SROW + col0 + n];
            d = __builtin_amdgcn_wmma_f32_16x16x4_f32(false, aop[q], false, b,
                                                      (short)0, d, false, false);
        }
        // all 32 lanes write useful rows: row = r + rowoff, col = col0 + n
#pragma unroll
        for (int r = 0; r < NB; ++r)
            Vsh[(r + rowoff) * SROW + col0 + n] = d[r];
    }
    __syncthreads();

    // ---- store: same contiguous pairing as the load ----
    float* dst = out + g0 * (long)DLEN;
#pragma unroll
    for (int r = 0; r < NB; ++r) {
#pragma unroll
        for (int j = 0; j < 2; ++j) {
            const int   f    = j * 256 + t;
            const int   col4 = f * 4;
            const int   row  = r + ((col4 >= DLEN) ? 8 : 0);
            const int   lcol = col4 & (DLEN - 1);
            const float4 v = *(const float4*)(&Vsh[row * SROW + lcol]);
            *(float4*)(dst + (long)r * NSTRIDE + col4) = v;
        }
    }
}

extern "C" void kernel_launch(void* const* d_in, const int* in_sizes, int n_in,
                              void* d_out, int out_size, void* d_ws, size_t ws_size,
                              hipStream_t stream)
{
    (void)in_sizes; (void)n_in; (void)d_ws; (void)ws_size; (void)out_size;
    const float* x   = (const float*)d_in[0];
    float*       out = (float*)d_out;
    // two (m,c) groups per block: 32*128/2 = 2048 blocks, 256 threads (8 waves)
    ortho_gs_wmma2_kernel<<<dim3(32 * 128 / 2), dim3(256), 0, stream>>>(x, out);
}